// TensorProduct_46540265620255
// MI455X (gfx1250) — compile-verified
//
#include <hip/hip_runtime.h>

// ---------------------------------------------------------------------------
// SO(3) tensor product, MI455X (gfx1250).
// Bandwidth-bound: 436 MB output stream @ 23.3 TB/s => ~19us floor.
// - CG (Wigner-3j-like) tables evaluated entirely at COMPILE TIME (constexpr
//   fp64 Racah formula + real<->complex basis change): coefficients become FMA
//   literals, zeros are pruned at compile time.
// - Inputs staged to LDS with the Tensor Data Mover (async tensor DMA) +
//   s_wait_tensorcnt.
// - One thread per (u,v) multiplicity pair; outer products reused across all
//   output l blocks of a pair; non-temporal streaming stores.
// ---------------------------------------------------------------------------

typedef unsigned int u32x4 __attribute__((ext_vector_type(4)));
typedef int          i32x8 __attribute__((ext_vector_type(8)));
typedef int          i32x4 __attribute__((ext_vector_type(4)));

// ===================== compile-time CG tables (fp64) =======================

constexpr int imax2(int a, int b){ return a > b ? a : b; }
constexpr int imin2(int a, int b){ return a < b ? a : b; }

constexpr double cfact(int n){
  double r = 1.0;
  for (int t = 2; t <= n; ++t) r *= (double)t;
  return r;
}

constexpr double csqrt_(double x){         // Newton, converges to fp64 ulp
  if (x <= 0.0) return 0.0;
  double g = x < 1.0 ? 1.0 : x;
  for (int i = 0; i < 48; ++i) g = 0.5*(g + x/g);
  return g;
}

struct cplx { double re, im; };
constexpr cplx cmul(cplx a, cplx b){ return { a.re*b.re - a.im*b.im, a.re*b.im + a.im*b.re }; }
constexpr cplx cconj(cplx a){ return { a.re, -a.im }; }

// Racah formula, transcribed from the reference
constexpr double cg_coeff(int j1,int m1,int j2,int m2,int j3,int m3){
  if (m3 != m1 + m2) return 0.0;
  const int vmin = imax2(imax2(-j1 + j2 + m3, -j1 + m1), 0);
  const int vmax = imin2(imin2(j2 + j3 + m1, j3 - j1 + j2), j3 + m3);
  const double C = csqrt_((2.0*j3 + 1.0) *
      (cfact(j3+j1-j2)*cfact(j3-j1+j2)*cfact(j1+j2-j3)*cfact(j3+m3)*cfact(j3-m3)) /
      (cfact(j1+j2+j3+1)*cfact(j1-m1)*cfact(j1+m1)*cfact(j2-m2)*cfact(j2+m2)));
  double S = 0.0;
  for (int v = vmin; v <= vmax; ++v){
    const double sgn = ((v + j2 + m2) & 1) ? -1.0 : 1.0;
    S += sgn * cfact(j2+j3+m1-v)*cfact(j1-m1+v) /
         (cfact(v)*cfact(j3-j1+j2-v)*cfact(j3+m3-v)*cfact(v+j1-j2-m3));
  }
  return C * S;
}

// q[a][b] of the real->complex change-of-basis matrix for given l (incl (-i)^l)
constexpr cplx qent(int l, int a, int b){
  const double inv = 0.70710678118654752440;
  const int m = a - l;
  cplx v = {0.0, 0.0};
  if (m == 0){
    if (b == l) v = {1.0, 0.0};
  } else if (m < 0){
    if      (b == l - m) v = { inv, 0.0 };
    else if (b == l + m) v = { 0.0, -inv };
  } else {
    const double s = (m & 1) ? -1.0 : 1.0;
    if      (b == l + m) v = { s*inv, 0.0 };
    else if (b == l - m) v = { 0.0, s*inv };
  }
  switch (l & 3){
    case 0:  return v;
    case 1:  return {  v.im, -v.re };   // * (-i)
    case 2:  return { -v.re, -v.im };   // * (-1)
    default: return { -v.im,  v.re };   // * (+i)
  }
}

template<int NO, int N1, int N2>
struct Tab { float v[NO][N1][N2]; };

template<int L1, int L2, int L>
constexpr Tab<2*L+1, 2*L1+1, 2*L2+1> make_cg(){
  constexpr int no = 2*L+1, n1 = 2*L1+1, n2 = 2*L2+1;
  const double inv_n = 1.0 / csqrt_(2.0*L2 + 1.0);   // 3j-like normalization
  const double outsc = csqrt_(2.0*L + 1.0);          // * sqrt(2l+1)
  double csu2[no][n1][n2] = {};
  for (int i = 0; i < no; ++i)
    for (int k = 0; k < n1; ++k)
      for (int n = 0; n < n2; ++n)
        csu2[i][k][n] = cg_coeff(L, i-L, L1, k-L1, L2, n-L2) * inv_n;
  Tab<no, n1, n2> t{};
  for (int j = 0; j < no; ++j)
    for (int lq = 0; lq < n1; ++lq)
      for (int mq = 0; mq < n2; ++mq){
        cplx acc = {0.0, 0.0};
        for (int i = 0; i < no; ++i){
          const cplx q1 = qent(L, i, j);
          if (q1.re == 0.0 && q1.im == 0.0) continue;
          for (int k = 0; k < n1; ++k){
            const cplx q2 = qent(L1, k, lq);
            if (q2.re == 0.0 && q2.im == 0.0) continue;
            const cplx q12 = cmul(q1, q2);
            for (int n = 0; n < n2; ++n){
              const double cv = csu2[i][k][n];
              if (cv == 0.0) continue;
              const cplx tt = cmul(q12, cconj(qent(L2, n, mq)));
              acc.re += tt.re * cv; acc.im += tt.im * cv;
            }
          }
        }
        t.v[j][lq][mq] = (float)(acc.re * outsc);
      }
  return t;
}

template<int L1, int L2, int L>
struct CGT {
  static constexpr Tab<2*L+1, 2*L1+1, 2*L2+1> t = make_cg<L1, L2, L>();
};
template<int L1, int L2, int L>
constexpr Tab<2*L+1, 2*L1+1, 2*L2+1> CGT<L1, L2, L>::t;

// ======================= TDM staging (async tensor DMA) ====================

// 1-row tile of n_f32 fp32 elements, global -> LDS, per ISA D# layout (cdna5 §8).
__device__ __forceinline__ void tdm_load_row(void* lds_dst, const void* gsrc, unsigned n_f32){
  unsigned lds = (unsigned)(size_t)lds_dst;                 // low 32 bits of flat LDS addr = LDS offset
  unsigned long long ga = (unsigned long long)(size_t)gsrc; // 57-bit global byte address
  u32x4 g0;
  g0.x = 1u;                                                // count=1, user descriptor
  g0.y = lds;                                               // lds_addr [63:32]
  g0.z = (unsigned)ga;                                      // global_addr lo
  g0.w = (unsigned)((ga >> 32) & 0x01FFFFFFull) | 0x80000000u; // ga hi + type=2 [127:126]
  i32x8 g1 = {
    0x00020000,                   // wg_mask=0, data_size=2 (4B)
    (int)(n_f32 << 16),           // tensor_dim0[15:0] at bits 63:48
    (int)(1u << 16),              // tensor_dim0 hi = 0, tensor_dim1 = 1
    (int)(n_f32 << 16),           // tensor_dim1 hi = 0, tile_dim0 = n
    1,                            // tile_dim1 = 1 (one row), tile_dim2 = 0
    (int)n_f32,                   // tensor_dim0_stride lo
    0, 0
  };
  i32x4 g2 = {1, 1, 0, 0};        // tensor_dim2=1, tensor_dim3=1, stride=0, tile_dim3=0
  i32x4 g3 = {0, 0, 0, 0};
  i32x8 g4 = {0, 0, 0, 0, 0, 0, 0, 0};  // extra SGPR block: unused, zeros
  __builtin_amdgcn_tensor_load_to_lds(g0, g1, g2, g3, g4, 0);
}

// ======================= main tensor-product kernel ========================

#define OUT_PITCH 13312

template<int D1, int D2>
__device__ __forceinline__ void outer_prod(const float* __restrict__ a,
                                           const float* __restrict__ b,
                                           float (&p)[D1][D2]){
#pragma unroll
  for (int j = 0; j < D1; ++j)
#pragma unroll
    for (int k = 0; k < D2; ++k)
      p[j][k] = a[j] * b[k];
}

// out[i] = sum_{j,k} C[i][j][k] * p[j][k]; zero coeffs pruned at compile time,
// nonzero coeffs are instruction literals.
template<int L1, int L2, int L>
__device__ __forceinline__ void emit_l(const float (&p)[2*L1+1][2*L2+1],
                                       float* __restrict__ op){
  constexpr int D1 = 2*L1+1, D2 = 2*L2+1, DO = 2*L+1;
#pragma unroll
  for (int i = 0; i < DO; ++i){
    float acc = 0.0f;
#pragma unroll
    for (int j = 0; j < D1; ++j)
#pragma unroll
      for (int k = 0; k < D2; ++k){
        const float c = CGT<L1, L2, L>::t.v[i][j][k];
        if (c != 0.0f) acc = fmaf(c, p[j][k], acc);
      }
    __builtin_nontemporal_store(acc, op + i);   // 436MB stream, written once
  }
}

__global__ __launch_bounds__(256)
void tp_kernel(const float* __restrict__ f1, const float* __restrict__ f2,
               float* __restrict__ out){
  __shared__ __align__(16) float sh_x1[208];
  __shared__ __align__(16) float sh_x2[64];

  const int z   = blockIdx.x;
  const int tid = threadIdx.x;

  // x1/x2 rows -> LDS via Tensor Data Mover (wave 0 issues, waits TENSORcnt)
  if (tid < 32){
    tdm_load_row(sh_x1, f1 + (size_t)z * 208, 208);
    tdm_load_row(sh_x2, f2 + (size_t)z * 64, 64);
    __builtin_amdgcn_s_wait_tensorcnt(0);
  }
  __syncthreads();

  float* oz = out + (size_t)z * OUT_PITCH;

  // ---- pairs with mul1=32 (512 (u,v) pairs): two rounds of 256 threads ----
#pragma unroll
  for (int half = 0; half < 2; ++half){
    const int t = tid + half * 256;
    const int u = t >> 4;          // 0..31
    const int v = t & 15;          // 0..15
    const float* a0 = sh_x1 + u;            // l1=0 slice (stride 1)
    const float* a1 = sh_x1 + 32 + u*3;     // l1=1 slice
    const float* b0 = sh_x2 + v;            // l2=0 slice
    const float* b1 = sh_x2 + 16 + v*3;     // l2=1 slice

    { float p[1][1]; outer_prod<1,1>(a0, b0, p); emit_l<0,0,0>(p, oz + 0    + t*1); }
    { float p[1][3]; outer_prod<1,3>(a0, b1, p); emit_l<0,1,1>(p, oz + 512  + t*3); }
    { float p[3][1]; outer_prod<3,1>(a1, b0, p); emit_l<1,0,1>(p, oz + 2048 + t*3); }
    { float p[3][3]; outer_prod<3,3>(a1, b1, p); // shared across l = 0,1,2
      emit_l<1,1,0>(p, oz + 3584 + t*1);
      emit_l<1,1,1>(p, oz + 4096 + t*3);
      emit_l<1,1,2>(p, oz + 5632 + t*5);
    }
  }

  // ---- pairs with mul1=16 (256 (u,v) pairs): one round ----
  {
    const int t = tid;
    const int u = t >> 4;          // 0..15
    const int v = t & 15;          // 0..15
    const float* a2 = sh_x1 + 128 + u*5;    // l1=2 slice
    const float* b0 = sh_x2 + v;
    const float* b1 = sh_x2 + 16 + v*3;

    { float p[5][1]; outer_prod<5,1>(a2, b0, p); emit_l<2,0,2>(p, oz + 8192 + t*5); }
    { float p[5][3]; outer_prod<5,3>(a2, b1, p); // shared across l = 1,2,3
      emit_l<2,1,1>(p, oz + 9472  + t*3);
      emit_l<2,1,2>(p, oz + 10240 + t*5);
      emit_l<2,1,3>(p, oz + 11520 + t*7);
    }
  }
}

// =============================== launch ====================================

extern "C" void kernel_launch(void* const* d_in, const int* in_sizes, int n_in,
                              void* d_out, int out_size, void* d_ws, size_t ws_size,
                              hipStream_t stream) {
  const float* f1 = (const float*)d_in[0];   // [B, 208] fp32
  const float* f2 = (const float*)d_in[1];   // [B, 64]  fp32
  const int B = in_sizes[0] / 208;

  tp_kernel<<<dim3(B), dim3(256), 0, stream>>>(f1, f2, (float*)d_out);
  (void)n_in; (void)out_size; (void)d_ws; (void)ws_size;
}